// MetaGAT_43490838839339
// MI455X (gfx1250) — compile-verified
//
#include <hip/hip_runtime.h>

// ---------------------------------------------------------------------------
// MetaGAT (5-layer GAT, H=2 heads, D=128, F=16) for MI455X / gfx1250.
//
// Refactored math (per layer):
//   q[h]    = ew_h^T @ att_j[h]           (16 floats/head, precomputed)
//   c[h]    = eb_h . att_j[h]
//   xw      = h @ W                        (WMMA f32 16x16x4 GEMM, LDS-staged B)
//   s_i[n,h]= xw[n,h,:] . att_i[h],  s_j[n,h] = xw[n,h,:] . att_j[h]
//   alpha   = lrelu( s_i[dst] + s_j[src] + edge_attr.q[h] + c[h] )
//   amax    = segment_max(alpha, dst)      (atomic u32 max, order-preserving key)
//   ea      = exp(alpha - amax[dst])
//   denom  += ea ;  aggr[dst] += ea*xw[src] ;  t[dst,h] += ea*edge_attr
//   out     = b + 0.5 * sum_h (aggr + t@ew_h^T + denom*eb_h) / (denom+EPS)
// ---------------------------------------------------------------------------

#define HEADS      2
#define D_FEAT     128
#define NN         256          // HEADS * D_FEAT
#define F_EDGE     16
#define NEG_SLOPE  0.2f
#define EPS_GAT    1e-16f
#define GEMM_ASYNC 1            // gfx1250 async-to-LDS staging of the W slab

typedef float v2f __attribute__((ext_vector_type(2)));
typedef float v8f __attribute__((ext_vector_type(8)));

// ---------------- float <-> order-preserving uint key (for atomic max) -----
__device__ __forceinline__ unsigned f2key(float f) {
    unsigned u = __float_as_uint(f);
    return (u & 0x80000000u) ? ~u : (u | 0x80000000u);
}
__device__ __forceinline__ float key2f(unsigned u) {
    return (u & 0x80000000u) ? __uint_as_float(u ^ 0x80000000u)
                             : __uint_as_float(~u);
}

// ---------------- zero accumulators ----------------------------------------
__global__ void k_zero(float4* __restrict__ p, int n4) {
    int i = blockIdx.x * blockDim.x + threadIdx.x;
    if (i < n4) p[i] = make_float4(0.f, 0.f, 0.f, 0.f);
}

// ---------------- per-layer tiny precompute: q[2][16], c[2] ----------------
__global__ void k_prep_qc(const float* __restrict__ ewl,   // [256][16]
                          const float* __restrict__ attl,  // [2][256]
                          const float* __restrict__ ebl,   // [256]
                          float* __restrict__ q,           // [32]
                          float* __restrict__ c) {         // [2]
    int t = threadIdx.x;
    if (t < 32) {
        int h = t >> 4, f = t & 15;
        const float* aj = attl + h * 2 * D_FEAT + D_FEAT;   // att_j[h]
        float acc = 0.f;
        for (int d = 0; d < D_FEAT; ++d)
            acc += ewl[(h * D_FEAT + d) * F_EDGE + f] * aj[d];
        q[t] = acc;
    } else if (t < 34) {
        int h = t - 32;
        const float* aj = attl + h * 2 * D_FEAT + D_FEAT;
        float acc = 0.f;
        for (int d = 0; d < D_FEAT; ++d)
            acc += ebl[h * D_FEAT + d] * aj[d];
        c[h] = acc;
    }
}

// ---------------- xw = X(Mx128) @ W(128x256) via V_WMMA_F32_16X16X4_F32 ----
// 256 threads = 8 waves. Block covers 8 M-tiles x one 64-col N-group.
// The 128x64 W slab (32 KB) is staged to LDS once per block with the
// gfx1250 async-to-LDS engine; each wave drains its own ASYNCcnt before
// the workgroup barrier (per CDNA5 ordering rules), then B comes from DS.
__global__ void k_gemm_xw(const float* __restrict__ X,
                          const float* __restrict__ Wl,   // [128][256] row-major
                          float* __restrict__ XW,
                          int Mtiles) {
    __shared__ float sW[D_FEAT * 64];                 // 32 KB of 320 KB WGP LDS
    const int nb = blockIdx.x & 3;                    // N-group (64 cols)
    const int mb = blockIdx.x >> 2;                   // group of 8 M-tiles
    const int n0 = nb * 64;

    // cooperative fill: 2048 float4, 8 per thread
    for (int i = threadIdx.x; i < D_FEAT * 16; i += 256) {
        const int k = i >> 4, cc = (i & 15) * 4;
        const float* gsrc = Wl + (size_t)k * NN + n0 + cc;
#if GEMM_ASYNC
        unsigned ldst = (unsigned)(size_t)(&sW[k * 64 + cc]);
        asm volatile("global_load_async_to_lds_b128 %0, %1, off"
                     :: "v"(ldst), "v"((unsigned long long)(size_t)gsrc)
                     : "memory");
#else
        *(float4*)(&sW[k * 64 + cc]) = *(const float4*)gsrc;
#endif
    }
#if GEMM_ASYNC
    asm volatile("s_wait_asynccnt 0x0" ::: "memory");  // own LDS writes landed
#endif
    __syncthreads();

    const int wave = threadIdx.x >> 5;
    const int lane = threadIdx.x & 31;
    const int mt   = mb * 8 + wave;
    if (mt >= Mtiles) return;                         // wave-uniform tail exit

    const int m0   = mt * 16;
    const int row  = lane & 15;          // A: M index / B: N index within tile
    const int koff = (lane >> 4) * 2;    // A/B: K pair selector (0 or 2)

    v8f acc[4] = {};   // 4 N-tiles of 16x16 f32

    for (int k0 = 0; k0 < D_FEAT; k0 += 4) {
        // A tile 16x4: lane holds X[m0+row][k0+koff .. +1]
        v2f a = *(const v2f*)(X + (size_t)(m0 + row) * D_FEAT + k0 + koff);
#pragma unroll
        for (int nt = 0; nt < 4; ++nt) {
            const int cn = nt * 16 + row;
            v2f b;
            b.x = sW[(k0 + koff    ) * 64 + cn];
            b.y = sW[(k0 + koff + 1) * 64 + cn];
            acc[nt] = __builtin_amdgcn_wmma_f32_16x16x4_f32(
                false, a, false, b, (short)0, acc[nt], false, false);
        }
    }

    // D layout: lanes 0-15 rows m0+i, lanes 16-31 rows m0+8+i
    const int rbase = m0 + (lane >> 4) * 8;
#pragma unroll
    for (int nt = 0; nt < 4; ++nt) {
        const int cn = n0 + nt * 16 + row;
#pragma unroll
        for (int i = 0; i < 8; ++i)
            XW[(size_t)(rbase + i) * NN + cn] = acc[nt][i];
    }
}

// ---------------- node scores s_i, s_j -------------------------------------
__global__ void k_scores(const float* __restrict__ XW,
                         const float* __restrict__ attl,  // [2][256]
                         float* __restrict__ si, float* __restrict__ sj,
                         int N) {
    int g = blockIdx.x * blockDim.x + threadIdx.x;
    if (g >= N * HEADS) return;
    int n = g >> 1, h = g & 1;
    const float4* x  = (const float4*)(XW + (size_t)n * NN + h * D_FEAT);
    const float4* ai = (const float4*)(attl + h * 2 * D_FEAT);
    const float4* aj = (const float4*)(attl + h * 2 * D_FEAT + D_FEAT);
    float a = 0.f, b = 0.f;
#pragma unroll 8
    for (int i = 0; i < D_FEAT / 4; ++i) {
        float4 xv = x[i], av = ai[i], bv = aj[i];
        a += xv.x * av.x + xv.y * av.y + xv.z * av.z + xv.w * av.w;
        b += xv.x * bv.x + xv.y * bv.y + xv.z * bv.z + xv.w * bv.w;
    }
    si[g] = a; sj[g] = b;
}

// ---------------- edge pass 1: logits + segment max ------------------------
__global__ void k_edge_alpha(const long long* __restrict__ ei,  // [2][E]
                             const float* __restrict__ eattr,   // [E][16]
                             const float* __restrict__ si,
                             const float* __restrict__ sj,
                             const float* __restrict__ q,       // [32]
                             const float* __restrict__ c,       // [2]
                             float* __restrict__ alpha,         // [E+N][2]
                             unsigned* __restrict__ amaxkey,    // [N][2]
                             int E, int EN) {
    int e = blockIdx.x * blockDim.x + threadIdx.x;
    if (e >= EN) return;
    int src, dst;
    float dot0 = 0.f, dot1 = 0.f;
    if (e < E) {
        src = (int)ei[e];
        dst = (int)ei[E + e];
        const float4* ea4 = (const float4*)(eattr + (size_t)e * F_EDGE);
        const float4* q0  = (const float4*)(q);
        const float4* q1  = (const float4*)(q + 16);
#pragma unroll
        for (int i = 0; i < 4; ++i) {
            float4 av = ea4[i], qa = q0[i], qb = q1[i];
            dot0 += av.x * qa.x + av.y * qa.y + av.z * qa.z + av.w * qa.w;
            dot1 += av.x * qb.x + av.y * qb.y + av.z * qb.z + av.w * qb.w;
        }
    } else {                         // self loop: edge_attr row is zero
        src = dst = e - E;
    }
    float a0 = si[dst * 2 + 0] + sj[src * 2 + 0] + dot0 + c[0];
    float a1 = si[dst * 2 + 1] + sj[src * 2 + 1] + dot1 + c[1];
    a0 = (a0 > 0.f) ? a0 : NEG_SLOPE * a0;   // leaky relu
    a1 = (a1 > 0.f) ? a1 : NEG_SLOPE * a1;
    alpha[e * 2 + 0] = a0;
    alpha[e * 2 + 1] = a1;
    atomicMax(&amaxkey[dst * 2 + 0], f2key(a0));
    atomicMax(&amaxkey[dst * 2 + 1], f2key(a1));
}

// ---------------- edge pass 2: exp + scatter (1 wave32 per edge) -----------
// e is forced wave-uniform with readfirstlane so the per-edge metadata
// (src/dst/alpha/amax) lowers to scalar SMEM loads instead of 32x VMEM.
__global__ void k_edge_scatter(const long long* __restrict__ ei,
                               const float* __restrict__ eattr,
                               const float* __restrict__ alpha,
                               const unsigned* __restrict__ amaxkey,
                               const float* __restrict__ XW,
                               float* __restrict__ aggr,    // [N][256]
                               float* __restrict__ t,       // [N][2][16]
                               float* __restrict__ denom,   // [N][2]
                               int E, int EN) {
    const int e = __builtin_amdgcn_readfirstlane(
                      (int)(blockIdx.x * 8 + (threadIdx.x >> 5)));
    if (e >= EN) return;
    const int lane = threadIdx.x & 31;

    int src, dst;
    if (e < E) {
        src = __builtin_amdgcn_readfirstlane((int)ei[e]);
        dst = __builtin_amdgcn_readfirstlane((int)ei[E + e]);
    } else {
        src = dst = e - E;
    }

    const float* xrow = XW + (size_t)src * NN + lane * 8;
    __builtin_prefetch(xrow, 0, 1);          // global_prefetch_b8: hide gather

    const float a0 = alpha[e * 2 + 0];
    const float a1 = alpha[e * 2 + 1];
    const float m0 = key2f(amaxkey[dst * 2 + 0]);
    const float m1 = key2f(amaxkey[dst * 2 + 1]);
    const float ea0 = __expf(a0 - m0);
    const float ea1 = __expf(a1 - m1);

    if (lane == 0) atomicAdd(&denom[dst * 2 + 0], ea0);
    if (lane == 1) atomicAdd(&denom[dst * 2 + 1], ea1);

    // coalesced 1KB row gather + weighted atomic scatter
    const float eah = (lane < 16) ? ea0 : ea1;   // cols 0..127 -> h0, 128..255 -> h1
    float4 p0 = *(const float4*)(xrow);
    float4 p1 = *(const float4*)(xrow + 4);
    float* arow = aggr + (size_t)dst * NN + lane * 8;
    atomicAdd(arow + 0, eah * p0.x);
    atomicAdd(arow + 1, eah * p0.y);
    atomicAdd(arow + 2, eah * p0.z);
    atomicAdd(arow + 3, eah * p0.w);
    atomicAdd(arow + 4, eah * p1.x);
    atomicAdd(arow + 5, eah * p1.y);
    atomicAdd(arow + 6, eah * p1.z);
    atomicAdd(arow + 7, eah * p1.w);

    if (e < E) {                 // factored edge_emb contribution (16/head)
        const int f = lane & 15;
        const float av = eattr[(size_t)e * F_EDGE + f];
        atomicAdd(&t[(size_t)dst * 32 + (lane >> 4) * 16 + f], eah * av);
    }
}

// ---------------- finalize: normalize, t@ew^T, bias, head-mean, relu -------
__global__ void k_finalize(const float* __restrict__ aggr,
                           const float* __restrict__ t,
                           const float* __restrict__ denom,
                           const float* __restrict__ ewl,   // [256][16]
                           const float* __restrict__ ebl,   // [256]
                           const float* __restrict__ bl,    // [128]
                           float* __restrict__ hout,        // [N][128]
                           int N, int do_relu) {
    __shared__ float sew[NN * F_EDGE];       // 16 KB of the 320 KB WGP LDS
    for (int i = threadIdx.x; i < NN * F_EDGE; i += blockDim.x)
        sew[i] = ewl[i];
    __syncthreads();

    int g = blockIdx.x * blockDim.x + threadIdx.x;
    if (g >= N * D_FEAT) return;
    int n = g >> 7, d = g & 127;

    float sum = 0.f;
#pragma unroll
    for (int h = 0; h < HEADS; ++h) {
        float den = denom[n * 2 + h];
        float inv = 1.f / (den + EPS_GAT);
        float acc = aggr[(size_t)n * NN + h * D_FEAT + d];
        const float* tv = t + (size_t)n * 32 + h * 16;
        const float* er = sew + (h * D_FEAT + d) * F_EDGE;
        float td = 0.f;
#pragma unroll
        for (int f = 0; f < F_EDGE; ++f) td += tv[f] * er[f];
        sum += (acc + td + den * ebl[h * D_FEAT + d]) * inv;
    }
    float v = 0.5f * sum + bl[d];
    if (do_relu) v = fmaxf(v, 0.f);
    hout[(size_t)n * D_FEAT + d] = v;
}

// ---------------------------------------------------------------------------
extern "C" void kernel_launch(void* const* d_in, const int* in_sizes, int n_in,
                              void* d_out, int out_size, void* d_ws, size_t ws_size,
                              hipStream_t stream) {
    const float*     x        = (const float*)d_in[0];
    const long long* eidx     = (const long long*)d_in[1];   // int64 per reference
    const float*     eattr    = (const float*)d_in[2];
    const float*     weights  = (const float*)d_in[3];
    const float*     atts     = (const float*)d_in[4];
    const float*     biases   = (const float*)d_in[5];
    const float*     edge_ws  = (const float*)d_in[6];
    const float*     edge_bs  = (const float*)d_in[7];
    float*           out      = (float*)d_out;

    const int N  = in_sizes[0] / D_FEAT;        // 10000
    const int E  = in_sizes[2] / F_EDGE;        // 320000
    const int EN = E + N;                       // with self loops

    // ---- workspace layout (floats) ----
    float*  ws      = (float*)d_ws;
    size_t  off     = 0;
    float*  aggr    = ws + off; off += (size_t)N * NN;      // zeroed
    float*  tbuf    = ws + off; off += (size_t)N * 32;      // zeroed
    float*  denom   = ws + off; off += (size_t)N * 2;       // zeroed
    unsigned* amaxk = (unsigned*)(ws + off); off += (size_t)N * 2;  // key 0 == -max
    const size_t zero_len = off;                            // contiguous zero region
    float*  xw      = ws + off; off += (size_t)N * NN;
    float*  si      = ws + off; off += (size_t)N * 2;
    float*  sj      = ws + off; off += (size_t)N * 2;
    float*  alpha   = ws + off; off += (size_t)EN * 2;
    float*  hA      = ws + off; off += (size_t)N * D_FEAT;
    float*  hB      = ws + off; off += (size_t)N * D_FEAT;
    float*  qbuf    = ws + off; off += 32;
    float*  cbuf    = ws + off; off += 2;

    const int z4       = (int)(zero_len / 4);
    const int Mtiles   = N / 16;                             // 625
    const int gemmBlks = ((Mtiles + 7) / 8) * 4;             // 8 M-tiles x 4 N-groups

    for (int l = 0; l < 5; ++l) {
        const float* Wl   = weights + (size_t)l * D_FEAT * NN;
        const float* attl = atts    + (size_t)l * HEADS * 2 * D_FEAT;
        const float* bl   = biases  + (size_t)l * D_FEAT;
        const float* ewl  = edge_ws + (size_t)l * NN * F_EDGE;
        const float* ebl  = edge_bs + (size_t)l * NN;

        const float* hin  = (l == 0) ? x : ((l & 1) ? hA : hB);
        float*       hout = (l == 4) ? out : ((l & 1) ? hB : hA);

        k_zero<<<(z4 + 255) / 256, 256, 0, stream>>>((float4*)ws, z4);
        k_prep_qc<<<1, 64, 0, stream>>>(ewl, attl, ebl, qbuf, cbuf);
        k_gemm_xw<<<gemmBlks, 256, 0, stream>>>(hin, Wl, xw, Mtiles);
        k_scores<<<(N * HEADS + 255) / 256, 256, 0, stream>>>(xw, attl, si, sj, N);
        k_edge_alpha<<<(EN + 255) / 256, 256, 0, stream>>>(
            eidx, eattr, si, sj, qbuf, cbuf, alpha, amaxk, E, EN);
        k_edge_scatter<<<(EN + 7) / 8, 256, 0, stream>>>(
            eidx, eattr, alpha, amaxk, xw, aggr, tbuf, denom, E, EN);
        k_finalize<<<(N * D_FEAT + 255) / 256, 256, 0, stream>>>(
            aggr, tbuf, denom, ewl, ebl, bl, hout, N, (l < 4) ? 1 : 0);
    }
    (void)n_in; (void)out_size; (void)ws_size;
}